// MPS_35012573397170
// MI455X (gfx1250) — compile-verified
//
#include <hip/hip_runtime.h>

typedef __attribute__((ext_vector_type(2))) float v2f;
typedef __attribute__((ext_vector_type(8))) float v8f;

#define DV      64          // bond dimension
#define PAD     68          // padded LDS row stride for P (floats)
#define MPAIR   80          // float2 stride per K-pair row of M (64 used + 16 pad)
#define NB      64          // batch
#define NCHUNK  8           // chunks per half-chain
#define CSITES  64          // sites per chunk (8*64 = 512 = half chain)

// ---------------------------------------------------------------------------
// Kernel 1: per (chain, batch, chunk) compute the ordered product of 64 site
// matrices  P = M_{s0} * M_{s0+1} * ... * M_{s0+63}  with fp32 WMMA.
// Grid: 1024 blocks, 128 threads (4 wave32). Each wave owns a 16-row slab of P.
// M is staged K-pair-interleaved so every B fragment is one ds_load_b64.
// ---------------------------------------------------------------------------
__global__ __launch_bounds__(128) void mps_chunk_kernel(
    const float* __restrict__ x,      // [64][1024][2]
    const float* __restrict__ Wl,     // [512][64][64][2]
    const float* __restrict__ Wr,     // [512][64][64][2]
    float* __restrict__ Pout)         // [2][64][8][64*64]
{
  __shared__ __align__(16) float2 mbuf2[32 * MPAIR];  // M, pair-interleaved
  __shared__ __align__(16) float  pb[2][DV * PAD];    // ping-pong running product P

  const int t    = threadIdx.x;
  const int lane = t & 31;
  const int wv   = t >> 5;

  const int bx    = blockIdx.x;
  const int chunk = bx & 7;
  const int b     = (bx >> 3) & 63;
  const int chain = bx >> 9;

  const float*  __restrict__ W  = chain ? Wr : Wl;
  const float2* __restrict__ W2 = reinterpret_cast<const float2*>(W);
  const float2* __restrict__ x2 = reinterpret_cast<const float2*>(x);
  const int s0 = chunk * CSITES;

  // Stage M[s] into the pair-interleaved B buffer:
  // mbuf2[p*MPAIR + r] = { M[2p][r], M[2p+1][r] }
  auto stageM = [&](int sc) {
    const float2 xv = x2[b * 1024 + chain * 512 + sc];
    const float2* wp = W2 + (size_t)sc * 4096;   // 64*64 float2 per site
    const int r = t & 63;
    const int h = t >> 6;
#pragma unroll
    for (int kk = 0; kk < 16; ++kk) {
      const int p = kk * 2 + h;                  // pair-row 0..31
      const float2 w0 = wp[(2 * p) * 64 + r];
      const float2 w1 = wp[(2 * p + 1) * 64 + r];
      float2 m;
      m.x = fmaf(w0.x, xv.x, w0.y * xv.y);
      m.y = fmaf(w1.x, xv.x, w1.y * xv.y);
      mbuf2[p * MPAIR + r] = m;
    }
  };

  // Stage M[s] row-major into a P buffer (chunk's first site: P starts as M)
  auto stageP = [&](int sc, float* dst) {
    const float2 xv = x2[b * 1024 + chain * 512 + sc];
    const float2* wp = W2 + (size_t)sc * 4096;
#pragma unroll
    for (int k = 0; k < 32; ++k) {
      const int e = k * 128 + t;                 // e = l*64 + r
      const float2 w2 = wp[e];
      dst[(e >> 6) * PAD + (e & 63)] = fmaf(w2.x, xv.x, w2.y * xv.y);
    }
  };

  stageP(s0, pb[0]);
  int cur = 0;

  // Fragment addressing (wave32, f32 16x16x4 layouts per ISA tables)
  const int half = (lane < 16) ? 0 : 2;   // K sub-pair selected by lane half
  const int n    = lane & 15;             // column within tile / row within A slab
  const int mrow = wv * 16 + n;           // A-matrix row for this lane

  for (int i = 1; i < CSITES; ++i) {
    stageM(s0 + i);
    if (i < CSITES - 1) {
      // prefetch next site's 32KB W slice (128 threads * 256B = 32KB)
      const float* nw = W + (size_t)(s0 + i + 1) * 8192;
      __builtin_prefetch(nw + t * 64, 0, 0);
    }
    __syncthreads();

    const float* __restrict__ pc = pb[cur];
    float* __restrict__ pn = pb[cur ^ 1];

    v8f acc[4] = {};   // 4 column tiles (16x16 each) of the new P slab

#pragma unroll
    for (int kk = 0; kk < 16; ++kk) {
      const int k0 = kk * 4 + half;
      // A 16x4 fragment from P: rows adjacent -> one b64 load (PAD even, k0 even)
      const v2f a = *reinterpret_cast<const v2f*>(&pc[mrow * PAD + k0]);
      const int bp = (k0 >> 1) * MPAIR + n;      // float2 index of {M[k0][n],M[k0+1][n]}
#pragma unroll
      for (int j = 0; j < 4; ++j) {
        // B 4x16 fragment from M: one b64 load, conflict-free banks
        const v2f bf = *reinterpret_cast<const v2f*>(&mbuf2[bp + j * 16]);
        acc[j] = __builtin_amdgcn_wmma_f32_16x16x4_f32(
            false, a, false, bf, (short)0, acc[j], false, false);
      }
    }

    // Spill new P slab to the ping-pong buffer (C/D layout -> [m][n])
#pragma unroll
    for (int r = 0; r < 8; ++r) {
      const int m = wv * 16 + r + ((lane < 16) ? 0 : 8);
#pragma unroll
      for (int j = 0; j < 4; ++j)
        pn[m * PAD + j * 16 + n] = acc[j][r];
    }
    cur ^= 1;
    __syncthreads();
  }

  // Write final chunk product to workspace, coalesced
  float* __restrict__ dst = Pout + (size_t)((chain * NB + b) * NCHUNK + chunk) * 4096;
  const float* __restrict__ src = pb[cur];
#pragma unroll
  for (int k = 0; k < 32; ++k) {
    const int e = k * 128 + t;
    dst[e] = src[(e >> 6) * PAD + (e & 63)];
  }
}

// ---------------------------------------------------------------------------
// Kernel 2: per batch, combine the 8 chunk products per chain with serial
// GEMVs against the boundary unit vector, then contract through out_core.
// Grid: 64 blocks, 64 threads.
// ---------------------------------------------------------------------------
__global__ __launch_bounds__(64) void mps_combine_kernel(
    const float* __restrict__ P,     // [2][64][8][4096]
    const float* __restrict__ oc,    // [10][64][64]
    float* __restrict__ out)         // [64][10]
{
  const int b = blockIdx.x;
  const int t = threadIdx.x;   // 0..63

  __shared__ float vbuf[DV];
  __shared__ float wbuf[DV];
  __shared__ float red[DV];

  // ---- left chain: vL^T = e^T P0 P1 ... P7  (v starts as row 0 of P0) ----
  const float* PL = P + (size_t)(0 * NB + b) * NCHUNK * 4096;
  float v = PL[t];                       // chunk0, row 0, col t
  for (int c = 1; c < NCHUNK; ++c) {
    vbuf[t] = v;
    __syncthreads();
    const float* Pc = PL + (size_t)c * 4096;
    float acc = 0.0f;
    for (int l = 0; l < DV; ++l) acc = fmaf(vbuf[l], Pc[l * DV + t], acc);
    v = acc;
    __syncthreads();
  }

  // ---- right chain: wR = P0 P1 ... P7 e  (w starts as col 0 of P7) ----
  const float* PR = P + (size_t)(1 * NB + b) * NCHUNK * 4096;
  float w = PR[7 * 4096 + t * DV];       // chunk7, row t, col 0
  for (int c = NCHUNK - 2; c >= 0; --c) {
    wbuf[t] = w;
    __syncthreads();
    const float* Pc = PR + (size_t)c * 4096;
    float acc = 0.0f;
    for (int r = 0; r < DV; ++r) acc = fmaf(Pc[t * DV + r], wbuf[r], acc);
    w = acc;
    __syncthreads();
  }
  wbuf[t] = w;
  __syncthreads();

  // ---- out[b,o] = sum_l v[l] * sum_r oc[o,l,r] * w[r] ----
  for (int o = 0; o < 10; ++o) {
    const float* ocp = oc + (size_t)(o * DV + t) * DV;
    float tmp = 0.0f;
    for (int r = 0; r < DV; ++r) tmp = fmaf(ocp[r], wbuf[r], tmp);
    red[t] = v * tmp;
    __syncthreads();
    if (t == 0) {
      float s = 0.0f;
      for (int l = 0; l < DV; ++l) s += red[l];
      out[b * 10 + o] = s;
    }
    __syncthreads();
  }
}

// ---------------------------------------------------------------------------
extern "C" void kernel_launch(void* const* d_in, const int* in_sizes, int n_in,
                              void* d_out, int out_size, void* d_ws, size_t ws_size,
                              hipStream_t stream) {
  (void)in_sizes; (void)n_in; (void)out_size; (void)ws_size;
  const float* x  = (const float*)d_in[0];   // [64,1024,2]
  const float* Wl = (const float*)d_in[1];   // [512,64,64,2]
  const float* oc = (const float*)d_in[2];   // [10,64,64]
  const float* Wr = (const float*)d_in[3];   // [512,64,64,2]
  float* out = (float*)d_out;                // [64,10]
  float* P   = (float*)d_ws;                 // 2*64*8*4096 floats = 16 MB

  mps_chunk_kernel<<<2 * NB * NCHUNK, 128, 0, stream>>>(x, Wl, Wr, P);
  mps_combine_kernel<<<NB, 64, 0, stream>>>(P, oc, out);
}